// FISTA_68582037782555
// MI455X (gfx1250) — compile-verified
//
#include <hip/hip_runtime.h>
#include <math.h>

typedef float v2f __attribute__((ext_vector_type(2)));
typedef float v4f __attribute__((ext_vector_type(4)));
typedef float v8f __attribute__((ext_vector_type(8)));

#define NP          256     // dictionary atoms
#define TDIM        4096    // time dimension
#define JDIM        128     // joints
#define BATCH       64
#define FISTA_ITERS 100
#define LAMBD       0.1f
#define RS          260     // transposed-C row stride (256 + 4) -> conflict-free b64 B loads

// ---------------------------------------------------------------------------
// Kernel 1: DtD = D^T D   (256x256, K=4096).  One wave per 16x16 output tile.
// ---------------------------------------------------------------------------
__global__ __launch_bounds__(32)
void fista_dtd_kernel(const float* __restrict__ D, float* __restrict__ DtD) {
    const int tile = blockIdx.x;          // 0..255
    const int mt = tile >> 4, nt = tile & 15;
    const int lane = threadIdx.x & 31;
    const int l15  = lane & 15;
    const int koff = (lane < 16) ? 0 : 2;
    const int mcol = mt * 16 + l15;       // A[m][k] = D[k][m]
    const int ncol = nt * 16 + l15;       // B[k][n] = D[k][n]
    v8f acc = {};
    for (int k = 0; k < TDIM; k += 4) {
        v2f a, b;
        a.x = D[(k + koff    ) * NP + mcol];
        a.y = D[(k + koff + 1) * NP + mcol];
        b.x = D[(k + koff    ) * NP + ncol];
        b.y = D[(k + koff + 1) * NP + ncol];
        acc = __builtin_amdgcn_wmma_f32_16x16x4_f32(false, a, false, b,
                                                    (short)0, acc, false, false);
    }
    const int mrow = mt * 16 + ((lane < 16) ? 0 : 8);
    #pragma unroll
    for (int r = 0; r < 8; ++r)
        DtD[(mrow + r) * NP + nt * 16 + l15] = acc[r];
}

// ---------------------------------------------------------------------------
// Kernel 2: power iteration -> L = 1 / spectral_norm(DtD).  One block.
// ---------------------------------------------------------------------------
__global__ __launch_bounds__(256)
void fista_power_kernel(const float* __restrict__ DtD, float* __restrict__ Lout) {
    __shared__ float v[NP];
    __shared__ float red[NP];
    const int tid = threadIdx.x;
    v[tid] = 1.0f;
    __syncthreads();
    float sigma = 1.0f;
    for (int it = 0; it < 64; ++it) {
        const float* row = DtD + tid * NP;
        float w = 0.0f;
        for (int p = 0; p < NP; ++p) w += row[p] * v[p];
        red[tid] = w * w;
        __syncthreads();
        for (int s = 128; s > 0; s >>= 1) {
            if (tid < s) red[tid] += red[tid + s];
            __syncthreads();
        }
        const float nrm = sqrtf(red[0]);
        sigma = nrm;
        v[tid] = w / nrm;
        __syncthreads();
    }
    if (tid == 0) Lout[0] = 1.0f / sigma;
}

// ---------------------------------------------------------------------------
// Kernel 3: DtY[b] = D^T @ Y[b]  (256x128, K=4096).  8 waves/block, 1 tile/wave.
// ---------------------------------------------------------------------------
__global__ __launch_bounds__(256)
void fista_dty_kernel(const float* __restrict__ D, const float* __restrict__ Y,
                      float* __restrict__ DtY) {
    const int b    = blockIdx.x >> 4;     // 64 batches
    const int grp  = blockIdx.x & 15;
    const int wave = threadIdx.x >> 5;
    const int lane = threadIdx.x & 31;
    const int t    = grp * 8 + wave;      // 0..127 tiles per batch
    const int mt = t >> 3, jt = t & 7;
    const int l15  = lane & 15;
    const int koff = (lane < 16) ? 0 : 2;
    const int mcol = mt * 16 + l15;
    const int jcol = jt * 16 + l15;
    const float* Yb = Y + (size_t)b * TDIM * JDIM;
    v8f acc = {};
    for (int k = 0; k < TDIM; k += 4) {
        v2f a, bv;
        a.x  = D [(k + koff    ) * NP   + mcol];
        a.y  = D [(k + koff + 1) * NP   + mcol];
        bv.x = Yb[(k + koff    ) * JDIM + jcol];
        bv.y = Yb[(k + koff + 1) * JDIM + jcol];
        acc = __builtin_amdgcn_wmma_f32_16x16x4_f32(false, a, false, bv,
                                                    (short)0, acc, false, false);
    }
    float* out = DtY + (size_t)b * NP * JDIM;
    const int mrow = mt * 16 + ((lane < 16) ? 0 : 8);
    #pragma unroll
    for (int r = 0; r < 8; ++r)
        out[(mrow + r) * JDIM + jt * 16 + l15] = acc[r];
}

// ---------------------------------------------------------------------------
// Kernel 4: 100 FISTA iterations, one block per batch, C resident in LDS.
//   - C stored TRANSPOSED: CT[col][row], stride 260 -> WMMA B operand is one
//     conflict-free ds_load_b64; elementwise phase is per-lane contiguous (b128)
//   - A operand: one global_load_b64 from row-major DtD (L0/L2 resident)
//   - DtY (loop-invariant) pre-scaled by L and preloaded into registers in
//     the WMMA accumulator layout
// ---------------------------------------------------------------------------
__global__ __launch_bounds__(256)
void fista_iter_kernel(const float* __restrict__ DtD,
                       const float* __restrict__ DtY,
                       const float* __restrict__ Lptr,
                       float* __restrict__ Cout) {
    extern __shared__ float lds[];
    float* CpT = lds;                 // C_prev^T [128][RS]
    float* CcT = lds + JDIM * RS;     // C_curr^T [128][RS]

    const int b    = blockIdx.x;
    const int tid  = threadIdx.x;
    const int wave = tid >> 5;
    const int lane = tid & 31;
    const int l15  = lane & 15;
    const int hi   = lane >> 4;               // 0 or 1
    const int koff = hi ? 2 : 0;

    const float* dty = DtY + (size_t)b * NP * JDIM;
    const float Lstep = Lptr[0];
    const float tau   = Lstep * LAMBD;

    // zero-init LDS
    for (int i = tid; i < 2 * JDIM * RS; i += 256) lds[i] = 0.0f;

    // preload L*DtY tiles into registers (WMMA accumulator layout, loop-invariant)
    v8f dreg[2][8];
    #pragma unroll
    for (int mi = 0; mi < 2; ++mi) {
        const int mrow0 = (wave * 2 + mi) * 16 + hi * 8;
        #pragma unroll
        for (int j = 0; j < 8; ++j) {
            const int col = j * 16 + l15;
            #pragma unroll
            for (int r = 0; r < 8; ++r)
                dreg[mi][j][r] = Lstep * dty[(mrow0 + r) * JDIM + col];
        }
    }
    __syncthreads();

    float t_prev = 1.0f;

    for (int it = 0; it < FISTA_ITERS; ++it) {
        v8f acc[2][8];
        for (int mi = 0; mi < 2; ++mi) {
            const int mt   = wave * 2 + mi;   // 16 m-tiles over 8 waves
            const int mcol = mt * 16 + l15;
            #pragma unroll
            for (int j = 0; j < 8; ++j) { v8f z = {}; acc[mi][j] = z; }
            for (int k = 0; k < NP; k += 4) {
                // A[m][k+koff], A[m][k+koff+1]: adjacent in row-major DtD
                const v2f a = *(const v2f*)&DtD[mcol * NP + k + koff];
                #pragma unroll
                for (int j = 0; j < 8; ++j) {
                    // B[k+koff][n], B[k+koff+1][n]: adjacent in transposed CT
                    const v2f bv = *(const v2f*)&CpT[(j * 16 + l15) * RS + k + koff];
                    acc[mi][j] = __builtin_amdgcn_wmma_f32_16x16x4_f32(
                        false, a, false, bv, (short)0, acc[mi][j], false, false);
                }
            }
        }

        const float t_next  = 0.5f * (1.0f + sqrtf(1.0f + 4.0f * t_prev * t_prev));
        const float t_const = (t_prev - 1.0f) / t_next;

        __syncthreads();   // all waves done reading C_prev

        for (int mi = 0; mi < 2; ++mi) {
            const int mrow0 = (wave * 2 + mi) * 16 + hi * 8;
            #pragma unroll
            for (int j = 0; j < 8; ++j) {
                const int base = (j * 16 + l15) * RS + mrow0;   // 8 contiguous rows
                float cp[8], cc[8], pn[8], cn[8];
                *(v4f*)&cp[0] = *(const v4f*)&CpT[base];
                *(v4f*)&cp[4] = *(const v4f*)&CpT[base + 4];
                *(v4f*)&cc[0] = *(const v4f*)&CcT[base];
                *(v4f*)&cc[4] = *(const v4f*)&CcT[base + 4];
                #pragma unroll
                for (int r = 0; r < 8; ++r) {
                    const float desc = cp[r] - Lstep * acc[mi][j][r] + dreg[mi][j][r];
                    const float mag  = fabsf(desc) - tau;
                    const float c    = (mag > 0.0f) ? copysignf(mag, desc) : 0.0f;
                    cn[r] = c;
                    pn[r] = c + t_const * (c - cc[r]);
                }
                *(v4f*)&CpT[base]     = *(const v4f*)&pn[0];
                *(v4f*)&CpT[base + 4] = *(const v4f*)&pn[4];
                *(v4f*)&CcT[base]     = *(const v4f*)&cn[0];
                *(v4f*)&CcT[base + 4] = *(const v4f*)&cn[4];
            }
        }
        t_prev = t_next;
        __syncthreads();
    }

    // write out C_curr (un-transpose: out[row*JDIM+col] = CcT[col*RS+row])
    float* out = Cout + (size_t)b * NP * JDIM;
    for (int i = tid; i < NP * JDIM; i += 256) {
        const int row = i >> 7, col = i & 127;
        out[i] = CcT[col * RS + row];
    }
}

// ---------------------------------------------------------------------------
extern "C" void kernel_launch(void* const* d_in, const int* in_sizes, int n_in,
                              void* d_out, int out_size, void* d_ws, size_t ws_size,
                              hipStream_t stream) {
    (void)in_sizes; (void)n_in; (void)out_size; (void)ws_size;
    const float* Y = (const float*)d_in[0];   // [64, 4096, 128]
    const float* D = (const float*)d_in[1];   // [4096, 256]
    float* out = (float*)d_out;               // [64, 256, 128]

    float* ws_f = (float*)d_ws;
    float* dtd  = ws_f;                        // 65536 floats (256 KB)
    float* Lp   = ws_f + 65536;                // 1 float
    float* dty  = ws_f + 65600;                // 2,097,152 floats (8 MB)

    const size_t fista_lds = (size_t)2 * JDIM * RS * sizeof(float); // 266,240 B
    hipFuncSetAttribute((const void*)fista_iter_kernel,
                        hipFuncAttributeMaxDynamicSharedMemorySize,
                        (int)fista_lds);

    // 1) DtD = D^T D
    fista_dtd_kernel<<<dim3(256), dim3(32), 0, stream>>>(D, dtd);
    // 2) L = 1 / spectral_norm(DtD)
    fista_power_kernel<<<dim3(1), dim3(256), 0, stream>>>(dtd, Lp);
    // 3) DtY[b] = D^T Y[b]
    fista_dty_kernel<<<dim3(BATCH * 16), dim3(256), 0, stream>>>(D, Y, dty);
    // 4) 100 FISTA iterations with C resident in LDS
    fista_iter_kernel<<<dim3(BATCH), dim3(256), fista_lds, stream>>>(dtd, dty, Lp, out);
}